// HeuristicBimodalCSRPool_2233382994383
// MI455X (gfx1250) — compile-verified
//
#include <hip/hip_runtime.h>
#include <hip/hip_bf16.h>
#include <cstdint>

#define DIM   128   // D
#define NFEAT 8     // F (FEAT = 0 is the pooled column)

typedef int   v4i_t __attribute__((ext_vector_type(4)));
typedef float v4f_t __attribute__((ext_vector_type(4)));
#define AS1 __attribute__((address_space(1)))
#define AS3 __attribute__((address_space(3)))

// CDNA5 async data path: global -> LDS async copy, tracked by ASYNCcnt.
// Toolchain declares the b128 builtin as (v4i AS1*, v4i AS3*, imm offset, imm cpol).
#if defined(__gfx1250__) && \
    __has_builtin(__builtin_amdgcn_global_load_async_to_lds_b128) && \
    __has_builtin(__builtin_amdgcn_s_wait_asynccnt)
#define USE_CDNA5_ASYNC 1
#else
#define USE_CDNA5_ASYNC 0
#endif

// Kernel 1: one thread per group. Sequential scan keeps the FIRST index that
// attains the strict maximum (matches segment_max + segment_min-of-candidates,
// including empty groups -> arg = N -> zeros row downstream).
__global__ __launch_bounds__(256) void group_argmax_kernel(
    const float* __restrict__ x_map,
    const int*   __restrict__ csr,
    int*         __restrict__ arg,
    int G, int N) {
  int g = blockIdx.x * blockDim.x + threadIdx.x;
  if (g >= G) return;
  int s = csr[g];
  int e = csr[g + 1];
  float best = -__builtin_inff();
  int   bi   = N;
  for (int i = s; i < e; ++i) {
    float v = x_map[i * NFEAT];   // column FEAT=0, stride F floats
    if (v > best) { best = v; bi = i; }
  }
  arg[g] = bi;
}

// Kernel 2: one wave32 per group. 32 lanes x b128 == one 512B row.
// All control flow is wave-uniform, so EXEC is all-ones inside active waves
// (required for the async-to-LDS path). Output is streamed with non-temporal
// stores so the 258 MB write-once result doesn't evict the gather working set
// from the 192 MB L2.
__global__ __launch_bounds__(256) void gather_row_kernel(
    const float* __restrict__ x_mod,
    const int*   __restrict__ arg,
    const int*   __restrict__ csr,
    float*       __restrict__ out_pool,
    float*       __restrict__ out_seen,
    int G, int N) {
#if USE_CDNA5_ASYNC
  __shared__ float tile[(256 / 32) * DIM];   // one row slot per wave in the block
#endif
  int wv   = (blockIdx.x * blockDim.x + threadIdx.x) >> 5;  // global wave id = group
  int lane = threadIdx.x & 31;
  if (wv >= G) return;

  int a = arg[wv];                                           // wave-uniform
  v4f_t v = {0.f, 0.f, 0.f, 0.f};
  if (a < N) {                                               // wave-uniform branch
#if USE_CDNA5_ASYNC
    float* row = &tile[(threadIdx.x >> 5) * DIM];
    const float* gsrc = x_mod + (size_t)a * DIM + lane * 4;
    __builtin_amdgcn_global_load_async_to_lds_b128(
        (AS1 v4i_t*)(void*)const_cast<float*>(gsrc),
        (AS3 v4i_t*)(void*)(row + lane * 4),
        /*offset=*/0, /*cpol=*/0);
    __builtin_amdgcn_s_wait_asynccnt(0);
    __asm__ volatile("" ::: "memory");   // keep the LDS read below the wait
    v = reinterpret_cast<const v4f_t*>(row)[lane];
#else
    v = reinterpret_cast<const v4f_t*>(x_mod + (size_t)a * DIM)[lane];
#endif
  }
  __builtin_nontemporal_store(
      v, reinterpret_cast<v4f_t*>(out_pool + (size_t)wv * DIM) + lane);

  if (lane == 0) {
    float seen = (csr[wv + 1] > csr[wv]) ? 1.0f : 0.0f;
    __builtin_nontemporal_store(seen, out_seen + wv);
  }
}

extern "C" void kernel_launch(void* const* d_in, const int* in_sizes, int n_in,
                              void* d_out, int out_size, void* d_ws, size_t ws_size,
                              hipStream_t stream) {
  // Inputs (setup_inputs order): x_main (unused), x_mod, x_map, csr_idx
  const float* x_mod = (const float*)d_in[1];
  const float* x_map = (const float*)d_in[2];
  const int*   csr   = (const int*)d_in[3];

  const int G = in_sizes[3] - 1;        // 500,000
  const int N = in_sizes[2] / NFEAT;    // 2,000,000

  int* arg = (int*)d_ws;                // G ints of scratch (2 MB)

  float* out_pool = (float*)d_out;                   // G x 128 floats
  float* out_seen = out_pool + (size_t)G * DIM;      // G floats (bool as 0/1)

  const int BLK = 256;
  group_argmax_kernel<<<dim3((G + BLK - 1) / BLK), dim3(BLK), 0, stream>>>(
      x_map, csr, arg, G, N);

  const int WAVES_PER_BLK = BLK / 32;   // wave32: 8 groups per block
  gather_row_kernel<<<dim3((G + WAVES_PER_BLK - 1) / WAVES_PER_BLK), dim3(BLK), 0, stream>>>(
      x_mod, arg, csr, out_pool, out_seen, G, N);
}